// EdgeEncoding_22737556865444
// MI455X (gfx1250) — compile-verified
//
#include <hip/hip_runtime.h>
#include <hip/hip_bf16.h>

// Problem constants (match reference: B, N, L, E, D)
#define B_ 16
#define N_ 512
#define L_ 5
#define E_ 2048
#define D_ 64

typedef __attribute__((ext_vector_type(2))) float v2f;
typedef __attribute__((ext_vector_type(8))) float v8f;

// ---------------------------------------------------------------------------
// Stage 1: scores[b][e][l] = dot(edge_embedding[b][e][:], edge_vector[l][:])
// One wave computes a 16(E-rows) x 16(L-cols, zero-padded past L=5) tile via
// 16 chained V_WMMA_F32_16X16X4_F32 (K=64 total), f32 accumulate — matches
// reference precision.
//
// Fragment layouts (ISA 7.12.2, 32-bit operands, wave32):
//   A (16x4, MxK): lanes 0-15: row=lane, {K=0,K=1}; lanes 16-31: row=lane-16,
//                  {K=2,K=3}  -> 2 VGPRs (v2f)
//   B (4x16, KxN): lanes 0-15: col=lane, {K=0,K=1}; lanes 16-31: col=lane-16,
//                  {K=2,K=3}  -> 2 VGPRs (v2f)
//   C/D (16x16):   VGPR r: lanes 0-15 -> row r, lanes 16-31 -> row r+8
// ---------------------------------------------------------------------------
__global__ __launch_bounds__(256) void scores_wmma_kernel(
    const float* __restrict__ emb,    // (B, E, D)
    const float* __restrict__ vec,    // (L, D)
    float* __restrict__ scores) {     // (B, E, L) f32 workspace
  const int wave = threadIdx.x >> 5;
  const int lane = threadIdx.x & 31;
  const int tile = blockIdx.x * (blockDim.x >> 5) + wave;
  const int tilesPerB = E_ / 16;                 // 128
  if (tile >= B_ * tilesPerB) return;            // wave-uniform guard
  const int b  = tile / tilesPerB;
  const int e0 = (tile - b * tilesPerB) * 16;

  const int mn = lane & 15;            // A: row within tile, B: column (=l)
  const int kh = (lane >> 4) << 1;     // 0 (lanes 0-15) or 2 (lanes 16-31)

  const float* Arow = emb + (((size_t)b * E_) + (size_t)(e0 + mn)) * D_ + kh;

  // Preload all 16 B-fragments (edge_vector is tiny; zero-pad cols 5..15).
  v2f bfrag[16];
#pragma unroll
  for (int s = 0; s < 16; ++s) {
    float bx = 0.0f, by = 0.0f;
    if (mn < L_) {
      const float* vrow = vec + mn * D_ + (s << 2) + kh;
      bx = vrow[0];
      by = vrow[1];
    }
    bfrag[s].x = bx;
    bfrag[s].y = by;
  }

  v8f acc = {};
#pragma unroll
  for (int s = 0; s < 16; ++s) {
    v2f a;
    a.x = Arow[(s << 2) + 0];
    a.y = Arow[(s << 2) + 1];
    // 8 args: (neg_a, A, neg_b, B, c_mod, C, reuse_a, reuse_b)
    acc = __builtin_amdgcn_wmma_f32_16x16x4_f32(false, a, false, bfrag[s],
                                                (short)0, acc, false, false);
  }

  // Write the 5 valid columns; each lane owns 8 rows of its column.
  if (mn < L_) {
    const int rbase = (lane < 16) ? 0 : 8;
#pragma unroll
    for (int r = 0; r < 8; ++r) {
      const int e = e0 + rbase + r;
      scores[(((size_t)b * E_) + (size_t)e) * L_ + mn] = acc[r];
    }
  }
}

// ---------------------------------------------------------------------------
// Stage 2: masked gather + mean. One thread per (b,n,m) output element.
// Streams 84 MB of indices exactly once -> non-temporal loads/stores so the
// 655 KB score table stays resident in cache for the random gathers.
// Consecutive threads read consecutive 20-byte index groups (fully coalesced:
// 32 lanes x 20 B = 640 B contiguous per wave).
// ---------------------------------------------------------------------------
__global__ __launch_bounds__(256) void gather_avg_kernel(
    const int* __restrict__ paths,    // (B, N, N, L)
    const float* __restrict__ scores, // (B, E, L)
    float* __restrict__ out) {        // (B, N, N)
  const size_t total = (size_t)B_ * N_ * N_;
  const size_t i = (size_t)blockIdx.x * blockDim.x + threadIdx.x;
  if (i >= total) return;

  const size_t b = i / ((size_t)N_ * N_);
  const float* __restrict__ sb = scores + b * (size_t)E_ * L_;
  const int* __restrict__ p = paths + i * L_;

  float sum = 0.0f;
  float cnt = 0.0f;
#pragma unroll
  for (int l = 0; l < L_; ++l) {
    const int e = __builtin_nontemporal_load(p + l);  // streamed once -> NT
    if ((unsigned)e < (unsigned)E_) {                 // e>=0 && e<=E-1
      sum += sb[(size_t)e * L_ + l];                  // hot table -> RT cached
      cnt += 1.0f;
    }
  }
  __builtin_nontemporal_store(sum / (cnt + 1e-9f), out + i);
}

extern "C" void kernel_launch(void* const* d_in, const int* in_sizes, int n_in,
                              void* d_out, int out_size, void* d_ws, size_t ws_size,
                              hipStream_t stream) {
  const float* emb   = (const float*)d_in[0];  // edge_embedding (B,E,D) f32
  const float* vec   = (const float*)d_in[1];  // edge_vector    (L,D)   f32
  const int*   paths = (const int*)d_in[2];    // edge_paths     (B,N,N,L) int
  float* scores = (float*)d_ws;                // (B,E,L) f32 = 655,360 bytes
  float* out    = (float*)d_out;               // (B,N,N) f32

  // Stage 1: 2048 tiles, 8 waves (256 threads) per block -> 256 blocks.
  const int tiles = B_ * (E_ / 16);
  const int wavesPerBlock = 8;
  scores_wmma_kernel<<<tiles / wavesPerBlock, 32 * wavesPerBlock, 0, stream>>>(
      emb, vec, scores);

  // Stage 2: one thread per output element (4.19M).
  const long long total = (long long)B_ * N_ * N_;
  const int bs = 256;
  gather_avg_kernel<<<(int)((total + bs - 1) / bs), bs, 0, stream>>>(
      paths, scores, out);
}